// GetSurroundingRegion1_16939351016273
// MI455X (gfx1250) — compile-verified
//
#include <hip/hip_runtime.h>
#include <math.h>

#define B_ 8
#define M_ 2048
#define C_ 256
#define H_ 240
#define W_ 240
#define HW_ (H_*W_)
#define STRIDE_ 9

typedef __attribute__((ext_vector_type(2))) float v2f;
typedef __attribute__((ext_vector_type(8))) float v8f;

// workspace layout (float offsets)
#define WS_AVG  0                      // B*C = 2048
#define WS_MX   2048                   // B*C
#define WS_S    4096                   // B*C
#define WS_CONF 6144                   // B*M = 16384
#define WS_INV  22528                  // B
#define WS_POOL 24576                  // B*M*C (16-byte aligned: 24576*4 = 98304)

// ---------------- Kernel 1: per-(b,c) mean & max over H*W ----------------
__global__ __launch_bounds__(256) void k_reduce(const float* __restrict__ seg,
                                                float* __restrict__ ws) {
  int blk = blockIdx.x;                               // b*C + c
  const float4* base = (const float4*)(seg + (size_t)blk * HW_);
  float s = 0.f, m = -INFINITY;
  for (int i = threadIdx.x; i < HW_ / 4; i += 256) {
    __builtin_prefetch((const char*)(base + i) + 8192, 0, 1);  // global_prefetch_b8
    float4 v = base[i];
    s += (v.x + v.y) + (v.z + v.w);
    m = fmaxf(m, fmaxf(fmaxf(v.x, v.y), fmaxf(v.z, v.w)));
  }
  __shared__ float ssum[256], smax[256];
  ssum[threadIdx.x] = s; smax[threadIdx.x] = m;
  __syncthreads();
  for (int off = 128; off > 0; off >>= 1) {
    if (threadIdx.x < off) {
      ssum[threadIdx.x] += ssum[threadIdx.x + off];
      smax[threadIdx.x] = fmaxf(smax[threadIdx.x], smax[threadIdx.x + off]);
    }
    __syncthreads();
  }
  if (threadIdx.x == 0) {
    ws[WS_AVG + blk] = ssum[0] * (1.0f / (float)HW_);
    ws[WS_MX  + blk] = smax[0];
  }
}

// ---------------- Kernel 2: channel-attention MLP via WMMA f32 16x16x4 ----
// Astack (16x256) = [avg(8x256); mx(8x256)].  H = relu(Astack @ W1^T) (16x16)
// out2 = H @ W2^T (16x256).  s[b][c] = sigmoid(out2[b][c] + out2[b+8][c]).
__global__ __launch_bounds__(32) void k_mlp_wmma(const float* __restrict__ w1,  // (16,256)
                                                 const float* __restrict__ w2,  // (256,16)
                                                 float* __restrict__ ws) {
  const int lane = threadIdx.x;
  const int n    = lane & 15;   // A-row index / B-col index
  const int half = lane >> 4;   // K sub-slot selector

  __shared__ float hlds[16 * 16];
  __shared__ float o2[16 * 256];

  const float* rowA = (n < 8) ? (ws + WS_AVG + n * C_)
                              : (ws + WS_MX + (n - 8) * C_);

  // GEMM1: 16x256 @ 256x16, K in steps of 4
  v8f acc = {};
  for (int k0 = 0; k0 < C_; k0 += 4) {
    int kb = k0 + 2 * half;
    v2f a; a.x = rowA[kb];            a.y = rowA[kb + 1];
    v2f b; b.x = w1[n * C_ + kb];     b.y = w1[n * C_ + kb + 1];  // B[k][n] = ca_w1[n][k]
    acc = __builtin_amdgcn_wmma_f32_16x16x4_f32(false, a, false, b,
                                                (short)0, acc, false, false);
  }
  // relu, store H in 16x16 row-major LDS (C-layout: lane holds [v+8*half][n])
#pragma unroll
  for (int v = 0; v < 8; ++v)
    hlds[(v + 8 * half) * 16 + n] = fmaxf(acc[v], 0.0f);
  __syncthreads();

  // GEMM2: 16x16 @ 16x256 -> 16 tiles of 16x16 along N, K=16 in 4 steps
  for (int nt = 0; nt < 16; ++nt) {
    v8f a2 = {};
#pragma unroll
    for (int st = 0; st < 4; ++st) {
      int kb = 4 * st + 2 * half;
      v2f a; a.x = hlds[n * 16 + kb];  a.y = hlds[n * 16 + kb + 1];  // A row = lane&15
      int col = nt * 16 + n;
      v2f b; b.x = w2[col * 16 + kb];  b.y = w2[col * 16 + kb + 1]; // B[k][col]=ca_w2[col][k]
      a2 = __builtin_amdgcn_wmma_f32_16x16x4_f32(false, a, false, b,
                                                 (short)0, a2, false, false);
    }
#pragma unroll
    for (int v = 0; v < 8; ++v)
      o2[(v + 8 * half) * 256 + nt * 16 + n] = a2[v];
  }
  __syncthreads();

  for (int i = lane; i < B_ * C_; i += 32) {
    int bb = i >> 8, c = i & 255;
    float z = o2[bb * 256 + c] + o2[(bb + 8) * 256 + c];
    ws[WS_S + i] = 1.0f / (1.0f + expf(-z));
  }
}

// ---------------- Kernel 3a: gather 9 taps, pool, center, per-kpt conf ----
__global__ __launch_bounds__(256) void k_gather(const float* __restrict__ kpts,
                                                const float* __restrict__ seg,
                                                const float* __restrict__ conv_w,
                                                const float* __restrict__ conv_b,
                                                const float* __restrict__ lin_w,
                                                const float* __restrict__ lin_b,
                                                float* __restrict__ out,
                                                float* __restrict__ ws) {
  int bm = blockIdx.x;                 // b*M + m
  int b  = bm >> 11;                   // M = 2048
  int c  = threadIdx.x;                // channel

  float kx = kpts[bm * 2 + 0];
  float ky = kpts[bm * 2 + 1];
  int x = (int)floorf(kx * (float)H_);
  int y = (int)floorf(ky * (float)W_);

  const float* plane = seg + (size_t)(b * C_ + c) * HW_;
  float pool = conv_b[0];
  float center = 0.f;
#pragma unroll
  for (int k = 0; k < 9; ++k) {
    int dx = STRIDE_ * (k / 3 - 1);
    int dy = STRIDE_ * (k % 3 - 1);
    int idx = (x + dx) * W_ + (y + dy);   // reference clips the FLAT index
    idx = min(max(idx, 0), HW_ - 1);
    float g = plane[idx];
    pool = fmaf(conv_w[k], g, pool);
    if (k == 4) center = g;
  }
  size_t base = (size_t)bm * C_;
  out[base + c] = center;               // center staged in d_out
  ws[WS_POOL + base + c] = pool;

  __shared__ float sd[256];
  sd[c] = center * lin_w[c];
  __syncthreads();
  for (int off = 128; off > 0; off >>= 1) {
    if (c < off) sd[c] += sd[c + off];
    __syncthreads();
  }
  if (c == 0) ws[WS_CONF + bm] = sd[0] + lin_b[0];
}

// ---------------- Kernel 3b: per-batch 1/denom over conf ------------------
__global__ __launch_bounds__(256) void k_denom(float* __restrict__ ws) {
  int b = blockIdx.x;
  float s = 0.f;
  for (int m = threadIdx.x; m < M_; m += 256) {
    float v = ws[WS_CONF + b * M_ + m];
    s += v * v;
  }
  __shared__ float sd[256];
  sd[threadIdx.x] = s;
  __syncthreads();
  for (int off = 128; off > 0; off >>= 1) {
    if (threadIdx.x < off) sd[threadIdx.x] += sd[threadIdx.x + off];
    __syncthreads();
  }
  if (threadIdx.x == 0)
    ws[WS_INV + b] = 1.0f / fmaxf(sqrtf(sd[0]), 1e-12f);
}

// ---------------- Kernel 3c: out = (1-conf)*pool + center + s -------------
__global__ __launch_bounds__(256) void k_final(float* __restrict__ out,
                                               const float* __restrict__ ws) {
  size_t f = (size_t)blockIdx.x * 256 + threadIdx.x;   // float4 index
  size_t e = f * 4;
  int    c  = (int)(e & (C_ - 1));
  size_t bm = e >> 8;
  int    b  = (int)(bm >> 11);

  float conf = ws[WS_CONF + bm] * ws[WS_INV + b];
  float w = 1.0f - conf;

  float4 p   = *(const float4*)(ws + WS_POOL + e);
  float4 ctr = *(float4*)(out + e);
  float4 sv  = *(const float4*)(ws + WS_S + (size_t)b * C_ + c);

  float4 r;
  r.x = fmaf(w, p.x, ctr.x + sv.x);
  r.y = fmaf(w, p.y, ctr.y + sv.y);
  r.z = fmaf(w, p.z, ctr.z + sv.z);
  r.w = fmaf(w, p.w, ctr.w + sv.w);
  *(float4*)(out + e) = r;
}

extern "C" void kernel_launch(void* const* d_in, const int* in_sizes, int n_in,
                              void* d_out, int out_size, void* d_ws, size_t ws_size,
                              hipStream_t stream) {
  const float* kpts   = (const float*)d_in[0];  // (B,M,2)
  const float* seg    = (const float*)d_in[1];  // (B,C,H,W)
  const float* conv_w = (const float*)d_in[2];  // (9,)
  const float* conv_b = (const float*)d_in[3];  // (1,)
  const float* lin_w  = (const float*)d_in[4];  // (C,)
  const float* lin_b  = (const float*)d_in[5];  // (1,)
  const float* ca_w1  = (const float*)d_in[6];  // (16,256)
  const float* ca_w2  = (const float*)d_in[7];  // (256,16)
  float* out = (float*)d_out;
  float* ws  = (float*)d_ws;

  // 1) streaming avg/max reduction over segment
  k_reduce<<<B_ * C_, 256, 0, stream>>>(seg, ws);
  // 2) channel-attention MLP on one wave via v_wmma_f32_16x16x4_f32
  k_mlp_wmma<<<1, 32, 0, stream>>>(ca_w1, ca_w2, ws);
  // 3) gather + pool + center + per-keypoint conf
  k_gather<<<B_ * M_, 256, 0, stream>>>(kpts, seg, conv_w, conv_b, lin_w, lin_b, out, ws);
  // 4) per-batch denom
  k_denom<<<B_, 256, 0, stream>>>(ws);
  // 5) fused finalize
  k_final<<<(B_ * M_ * C_) / (256 * 4), 256, 0, stream>>>(out, ws);
}